// MultiHeadAttention_14353780703743
// MI455X (gfx1250) — compile-verified
//
#include <hip/hip_runtime.h>
#include <hip/hip_bf16.h>
#include <stdint.h>

typedef __attribute__((ext_vector_type(2))) float v2f;
typedef __attribute__((ext_vector_type(8))) float v8f;

#define DIMC   512
#define NHEAD  8
#define EHEAD  64
#define SEQL   1024
#define NBATCH 4
#define NBUCK  1023
#define MAXT   512
#define VSTR   72   // V chunk LDS row stride (floats): rows kk / kk+2 hit disjoint bank groups

__device__ __forceinline__ v8f wmma4(v2f a, v2f b, v8f c) {
  // D(16x16,f32) = A(16x4,f32) x B(4x16,f32) + C
  return __builtin_amdgcn_wmma_f32_16x16x4_f32(false, a, false, b, (short)0, c, false, false);
}

// C[M,N] = A[M,K] @ B[N,K]^T + bias[N]   (one wave per 32x32 tile: 4 loads : 4 WMMAs)
__global__ void __launch_bounds__(256) gemm_tn_bias(
    const float* __restrict__ A, const float* __restrict__ B,
    const float* __restrict__ bias, float* __restrict__ C,
    int M, int N, int K) {
  int w    = threadIdx.x >> 5;
  int lane = threadIdx.x & 31;
  int half = lane >> 4;
  int lm   = lane & 15;
  int wid  = blockIdx.x * (blockDim.x >> 5) + w;
  int tiles_n = N >> 5;
  int tm = wid / tiles_n, tn = wid % tiles_n;

  const float* a0 = A + (size_t)(tm * 32 + lm) * K + 2 * half;
  const float* a1 = a0 + (size_t)16 * K;
  const float* b0 = B + (size_t)(tn * 32 + lm) * K + 2 * half;
  const float* b1 = b0 + (size_t)16 * K;
  v8f acc00 = {}, acc01 = {}, acc10 = {}, acc11 = {};
  for (int k = 0; k < K; k += 4) {
    v2f A0 = *(const v2f*)(a0 + k);
    v2f A1 = *(const v2f*)(a1 + k);
    v2f B0 = *(const v2f*)(b0 + k);
    v2f B1 = *(const v2f*)(b1 + k);
    acc00 = wmma4(A0, B0, acc00);
    acc01 = wmma4(A0, B1, acc01);
    acc10 = wmma4(A1, B0, acc10);
    acc11 = wmma4(A1, B1, acc11);
  }
  int col0 = tn * 32 + lm, col1 = col0 + 16;
  float bn0 = bias[col0], bn1 = bias[col1];
#pragma unroll
  for (int r = 0; r < 8; ++r) {
    int row0 = tm * 32 + r + 8 * half;
    int row1 = row0 + 16;
    C[(size_t)row0 * N + col0] = acc00[r] + bn0;
    C[(size_t)row0 * N + col1] = acc01[r] + bn1;
    C[(size_t)row1 * N + col0] = acc10[r] + bn0;
    C[(size_t)row1 * N + col1] = acc11[r] + bn1;
  }
}

// qkv[b,l, q|k|v (DIM each)] -> rope(q),rope(k),v in [b,h,l,e] layout
__global__ void __launch_bounds__(256) rope_split_kernel(
    const float* __restrict__ qkv, const int* __restrict__ time_ids,
    float* __restrict__ q_r, float* __restrict__ k_r, float* __restrict__ v_r) {
  int idx = blockIdx.x * blockDim.x + threadIdx.x;
  int j = idx & 31;            // freq index (half = 32)
  int h = (idx >> 5) & 7;
  int l = (idx >> 8) & 1023;
  int b = idx >> 18;
  float t = (float)time_ids[b * SEQL + l];
  float invf = powf(10000.0f, -(float)j * (1.0f / 32.0f));
  float sn, cs;
  sincosf(t * invf, &sn, &cs);
  const float* base = qkv + (size_t)(b * SEQL + l) * (3 * DIMC);
  size_t ob = ((size_t)(b * NHEAD + h) * SEQL + l) * EHEAD;
  float q1 = base[h * EHEAD + j], q2 = base[h * EHEAD + j + 32];
  q_r[ob + j]      = q1 * cs - q2 * sn;
  q_r[ob + j + 32] = q2 * cs + q1 * sn;
  float k1 = base[DIMC + h * EHEAD + j], k2 = base[DIMC + h * EHEAD + j + 32];
  k_r[ob + j]      = k1 * cs - k2 * sn;
  k_r[ob + j + 32] = k2 * cs + k1 * sn;
  v_r[ob + j]      = base[2 * DIMC + h * EHEAD + j];
  v_r[ob + j + 32] = base[2 * DIMC + h * EHEAD + j + 32];
}

// Flash attention: 4 waves/block share one (b,h); each wave owns a 16-row q tile.
// K/V chunks are staged in LDS via async-to-LDS (double buffered, ASYNCcnt).
__global__ void __launch_bounds__(128) attn_kernel(
    const float* __restrict__ q_r, const float* __restrict__ k_r,
    const float* __restrict__ v_r, const int* __restrict__ time_ids,
    const uint8_t* __restrict__ mask, const float* __restrict__ rel_table,
    const float* __restrict__ post_table, float* __restrict__ attn_out) {
  // K buffer is pre-swizzled into B-fragment order: word = 32*(2*st+half) + 2*row + d
  __shared__ __align__(16) float ldsK[2][16 * EHEAD];
  __shared__ __align__(16) float ldsV[2][16 * VSTR];
  __shared__ float ldsP[4][256];

  int tid  = threadIdx.x;
  int w    = tid >> 5;
  int lane = tid & 31;
  int half = lane >> 4;
  int lm   = lane & 15;
  float* myp = &ldsP[w][0];
  int wid = blockIdx.x * 4 + w;
  int qt = wid & 63;
  int h  = (wid >> 6) & 7;
  int b  = wid >> 9;

  const float* qb = q_r + ((size_t)(b * NHEAD + h) * SEQL + qt * 16) * EHEAD;
  const float* kb = k_r + (size_t)(b * NHEAD + h) * SEQL * EHEAD;
  const float* vb = v_r + (size_t)(b * NHEAD + h) * SEQL * EHEAD;
  const int* T = time_ids + b * SEQL;
  const uint8_t* Mk = mask + b * SEQL;
  const float* rel  = rel_table + h * NBUCK;
  const float* post = post_table + h * NBUCK;

  // async copy of chunk kc into buffer buf: K as swizzled b64, V as row-major b128
  auto issue_chunk = [&](int buf, int kc) {
    const float* kg = kb + (size_t)kc * 16 * EHEAD;
    const float* vg = vb + (size_t)kc * 16 * EHEAD;
    uint32_t kds = (uint32_t)(uintptr_t)&ldsK[buf][0];
    uint32_t vds = (uint32_t)(uintptr_t)&ldsV[buf][0];
#pragma unroll
    for (int j = 0; j < 4; ++j) {           // 512 b64 transfers for K
      int i = tid + 128 * j;                // i = sh*16 + r, sh = 2*st+half
      int r = i & 15;
      int sh = i >> 4;
      uint64_t ga = (uint64_t)(uintptr_t)(kg + r * EHEAD + sh * 2);
      uint32_t la = kds + (uint32_t)(sh * 32 + r * 2) * 4;
      asm volatile("global_load_async_to_lds_b64 %0, %1, off"
                   :: "v"(la), "v"(ga) : "memory");
    }
#pragma unroll
    for (int j = 0; j < 2; ++j) {           // 256 b128 transfers for V
      int i = tid + 128 * j;                // i = r*16 + seg
      int r = i >> 4;
      int seg = i & 15;
      uint64_t ga = (uint64_t)(uintptr_t)(vg + r * EHEAD + seg * 4);
      uint32_t la = vds + (uint32_t)(r * VSTR + seg * 4) * 4;
      asm volatile("global_load_async_to_lds_b128 %0, %1, off"
                   :: "v"(la), "v"(ga) : "memory");
    }
  };

  // q tile pre-loaded in A layout: lane lm holds row lm, k-pair (4s + 2*half)
  v2f qa[16];
#pragma unroll
  for (int s = 0; s < 16; ++s)
    qa[s] = *(const v2f*)(qb + (size_t)lm * EHEAD + 4 * s + 2 * half);

  int tqm[8];
#pragma unroll
  for (int r = 0; r < 8; ++r) tqm[r] = T[qt * 16 + r + 8 * half];
  int tqa = T[qt * 16 + lm];

  float mi[8], li[8];
  v8f zero = {};
  v8f acc_o[4], acc_p[4];
#pragma unroll
  for (int r = 0; r < 8; ++r) { mi[r] = -1e30f; li[r] = 0.0f; }
#pragma unroll
  for (int g = 0; g < 4; ++g) { acc_o[g] = zero; acc_p[g] = zero; }

  const float scale = 0.125f;  // 1/sqrt(64)

  issue_chunk(0, 0);
  for (int kc = 0; kc < 64; ++kc) {
    int cur = kc & 1;
    asm volatile("s_wait_asynccnt 0x0" ::: "memory");
    __syncthreads();                         // chunk kc resident for all waves
    if (kc < 63) issue_chunk(cur ^ 1, kc + 1);

    const float* Kc = &ldsK[cur][0];
    const float* Vc = &ldsV[cur][0];
    int col = kc * 16 + lm;
    int tk = T[col];
    bool mv = Mk[col] != 0;

    // S = Q @ K^T  (B operand from swizzled LDS: conflict-free ds_load_b64)
    v8f s = zero;
#pragma unroll
    for (int st = 0; st < 16; ++st) {
      v2f bk = *(const v2f*)(Kc + 32 * (2 * st + half) + 2 * lm);
      s = wmma4(qa[st], bk, s);
    }

    // bias + mask + online softmax in D layout (row r+8*half, col = lane)
    float p[8];
#pragma unroll
    for (int r = 0; r < 8; ++r) {
      int idx = tqm[r] - tk + (MAXT - 1);
      idx = idx < 0 ? 0 : (idx > NBUCK - 1 ? NBUCK - 1 : idx);
      float sv = mv ? (s[r] * scale + rel[idx]) : -1e30f;
      float rmax = sv;
      rmax = fmaxf(rmax, __shfl_xor(rmax, 1));
      rmax = fmaxf(rmax, __shfl_xor(rmax, 2));
      rmax = fmaxf(rmax, __shfl_xor(rmax, 4));
      rmax = fmaxf(rmax, __shfl_xor(rmax, 8));
      float mnew  = fmaxf(mi[r], rmax);
      float alpha = __expf(mi[r] - mnew);
      float pv    = mv ? __expf(sv - mnew) : 0.0f;
      float rsum = pv;
      rsum += __shfl_xor(rsum, 1);
      rsum += __shfl_xor(rsum, 2);
      rsum += __shfl_xor(rsum, 4);
      rsum += __shfl_xor(rsum, 8);
      li[r] = li[r] * alpha + rsum;
      mi[r] = mnew;
      p[r] = pv;
#pragma unroll
      for (int g = 0; g < 4; ++g) acc_o[g][r] *= alpha;
    }

    // P: D layout -> row-major LDS (per-wave slice) for A-layout re-read
#pragma unroll
    for (int r = 0; r < 8; ++r)
      myp[(r + 8 * half) * 16 + lm] = p[r];
    __syncthreads();

    // acc_o += P @ Vchunk ; acc_p += PB @ Vchunk  (shared B operand from LDS)
#pragma unroll
    for (int st = 0; st < 4; ++st) {
      int kk = 4 * st + 2 * half;
      v2f pa = { myp[lm * 16 + kk], myp[lm * 16 + kk + 1] };
      int kcol = kc * 16 + kk;
      int t0 = T[kcol], t1 = T[kcol + 1];
      bool m0 = Mk[kcol] != 0, m1 = Mk[kcol + 1] != 0;
      int i0 = tqa - t0 + (MAXT - 1); i0 = i0 < 0 ? 0 : (i0 > NBUCK - 1 ? NBUCK - 1 : i0);
      int i1 = tqa - t1 + (MAXT - 1); i1 = i1 < 0 ? 0 : (i1 > NBUCK - 1 ? NBUCK - 1 : i1);
      v2f pb = { m0 ? post[i0] : 0.0f, m1 ? post[i1] : 0.0f };
      const float* vr0 = Vc + kk * VSTR;
#pragma unroll
      for (int g = 0; g < 4; ++g) {
        v2f bvv = { vr0[16 * g + lm], vr0[VSTR + 16 * g + lm] };
        acc_o[g] = wmma4(pa, bvv, acc_o[g]);
        acc_p[g] = wmma4(pb, bvv, acc_p[g]);
      }
    }
    __syncthreads();                         // readers done before buffer reuse
  }

  // store in [b, l, h*64+e] so the out-projection is a plain GEMM
#pragma unroll
  for (int r = 0; r < 8; ++r) {
    float inv = li[r] > 0.0f ? 1.0f / li[r] : 0.0f;
    int row = qt * 16 + r + 8 * half;
    float* op = attn_out + ((size_t)(b * SEQL) + row) * DIMC + h * EHEAD;
#pragma unroll
    for (int g = 0; g < 4; ++g)
      op[16 * g + lm] = acc_o[g][r] * inv + acc_p[g][r];
  }
}

extern "C" void kernel_launch(void* const* d_in, const int* in_sizes, int n_in,
                              void* d_out, int out_size, void* d_ws, size_t ws_size,
                              hipStream_t stream) {
  const float*   x         = (const float*)d_in[0];
  const uint8_t* mask      = (const uint8_t*)d_in[1];
  const int*     time_ids  = (const int*)d_in[2];
  const float*   W_in      = (const float*)d_in[3];
  const float*   b_in      = (const float*)d_in[4];
  const float*   W_out     = (const float*)d_in[5];
  const float*   b_out     = (const float*)d_in[6];
  const float*   rel_tab   = (const float*)d_in[7];
  const float*   post_tab  = (const float*)d_in[8];
  float* out = (float*)d_out;
  float* ws  = (float*)d_ws;

  float* qkv = ws;                                        // 4*1024*1536 floats
  float* q_r = ws + (size_t)NBATCH * SEQL * 3 * DIMC;     // 4*8*1024*64 each
  float* k_r = q_r + (size_t)NBATCH * NHEAD * SEQL * EHEAD;
  float* v_r = k_r + (size_t)NBATCH * NHEAD * SEQL * EHEAD;
  float* attn = ws;                                       // alias qkv (dead after rope)

  // qkv = x @ W_in^T + b_in : M=4096, N=1536, K=512 -> 6144 32x32 tiles / 8 waves
  gemm_tn_bias<<<768, 256, 0, stream>>>(x, W_in, b_in, qkv, NBATCH * SEQL, 3 * DIMC, DIMC);
  // rope + head-major split
  rope_split_kernel<<<4096, 256, 0, stream>>>(qkv, time_ids, q_r, k_r, v_r);
  // flash attention with rel-bias softmax + post-bias matmul, async K/V staging
  attn_kernel<<<512, 128, 0, stream>>>(q_r, k_r, v_r, time_ids, mask, rel_tab, post_tab, attn);
  // out = attn @ W_out^T + b_out : M=4096, N=512, K=512 -> 2048 tiles
  gemm_tn_bias<<<256, 256, 0, stream>>>(attn, W_out, b_out, out, NBATCH * SEQL, DIMC, DIMC);
}